// EncoderLayer_63874753626789
// MI455X (gfx1250) — compile-verified
//
#include <hip/hip_runtime.h>
#include <math.h>

// ---------------------------------------------------------------------------
// Problem constants (from the reference)
// ---------------------------------------------------------------------------
#define LEN     190            // N_CODE + PC
#define BATCH   16
#define DMODEL  512
#define DINNER  1024
#define DSTATE  16
#define DTRANK  32
#define B2      32             // fwd + reversed sequences batched together
#define ROWS    (BATCH * LEN)  // 3040
#define ROWS2   (B2 * LEN)     // 6080

typedef float v2f __attribute__((ext_vector_type(2)));
typedef float v8f __attribute__((ext_vector_type(8)));

__device__ __forceinline__ float silu_f(float v) { return v / (1.0f + expf(-v)); }

// ---------------------------------------------------------------------------
// Epilogue applied at GEMM store time
//   EPI 0: none
//   EPI 1: += bias[col]
//   EPI 2: (acc + bias[col]) * aux[row,col]
//   EPI 3: (acc + bias[col]) * (mask[row % maskL, col] != 0)
//   EPI 4: softplus(acc + bias[col])
// ---------------------------------------------------------------------------
template <int EPI>
__device__ __forceinline__ float epi_apply(float val, int rr, int cc, int ldc, int N,
                                           const float* __restrict__ bias,
                                           const float* __restrict__ aux,
                                           const unsigned char* __restrict__ mask,
                                           int maskL)
{
    if (EPI == 1) {
        val += bias[cc];
    } else if (EPI == 2) {
        val = (val + bias[cc]) * aux[(size_t)rr * ldc + cc];
    } else if (EPI == 3) {
        val = (val + bias[cc]) * (mask[(size_t)(rr % maskL) * N + cc] ? 1.0f : 0.0f);
    } else if (EPI == 4) {
        val += bias[cc];
        val = (val > 20.0f) ? val : log1pf(expf(val));
    }
    return val;
}

// ---------------------------------------------------------------------------
// WMMA fp32 GEMM:  C(M,N) = A(M,K) @ W(N,K)^T  (+ fused epilogue)
// 4 waves / block; each wave owns a 16x32 C tile (two accumulators sharing
// one A fragment) -> block tile 32(M) x 64(N).
// Per the CDNA5 32-bit A 16x4 layout, each lane's A fragment is the
// contiguous pair A[row][k0+2*half .. +1]; the B(4x16) fragment mirrors it
// from W[col][k0+2*half .. +1], so all fragments are plain float2 loads.
// K is always even here; main loop is branch-free, the K%4==2 tail zeroes
// the upper-half A pair (zero A kills the out-of-range k contributions).
// ---------------------------------------------------------------------------
template <int EPI>
__global__ __launch_bounds__(128) void wmma_gemm_nt(
    const float* __restrict__ A, int lda,
    const float* __restrict__ W, int ldw,
    float* __restrict__ C, int ldc,
    int M, int N, int K,
    const float* __restrict__ bias,
    const float* __restrict__ aux,
    const unsigned char* __restrict__ mask, int maskL)
{
    const int lane  = threadIdx.x & 31;
    const int wave  = threadIdx.x >> 5;
    const int half  = lane >> 4;     // 0: lanes 0-15, 1: lanes 16-31
    const int r     = lane & 15;
    const int tileM = blockIdx.y * 32 + (wave >> 1) * 16;
    const int tileN = blockIdx.x * 64 + (wave & 1) * 32;
    if (tileM >= M || tileN >= N) return;   // wave-uniform exit

    const int row = tileM + r;
    const int c0  = tileN + r;
    const int c1  = tileN + 16 + r;
    // Clamp OOB rows/cols to a valid address: garbage in A row m only feeds
    // C row m (never stored), likewise for B columns -> safe with full EXEC.
    const float* arow = A + (size_t)((row < M) ? row : (M - 1)) * lda + 2 * half;
    const float* wr0  = W + (size_t)((c0  < N) ? c0  : (N - 1)) * ldw + 2 * half;
    const float* wr1  = W + (size_t)((c1  < N) ? c1  : (N - 1)) * ldw + 2 * half;

    v8f acc0 = {};
    v8f acc1 = {};
    const int K4 = K & ~3;
#pragma unroll 4
    for (int k0 = 0; k0 < K4; k0 += 4) {
        const v2f av  = *(const v2f*)(arow + k0);
        const v2f bv0 = *(const v2f*)(wr0 + k0);
        const v2f bv1 = *(const v2f*)(wr1 + k0);
        acc0 = __builtin_amdgcn_wmma_f32_16x16x4_f32(false, av, false, bv0,
                                                     (short)0, acc0, false, false);
        acc1 = __builtin_amdgcn_wmma_f32_16x16x4_f32(false, av, false, bv1,
                                                     (short)0, acc1, false, false);
    }
    if (K & 2) {
        // Lower half (half==0) covers k = K4, K4+1 (real); upper half would be
        // k = K4+2, K4+3 (OOB) -> zero its A pair. Loads stay in range: read the
        // last valid pair for both halves.
        const float g = (half == 0) ? 1.0f : 0.0f;
        v2f av  = *(const v2f*)(arow - 2 * half + K4);
        v2f bv0 = *(const v2f*)(wr0  - 2 * half + K4);
        v2f bv1 = *(const v2f*)(wr1  - 2 * half + K4);
        av.x *= g; av.y *= g;
        acc0 = __builtin_amdgcn_wmma_f32_16x16x4_f32(false, av, false, bv0,
                                                     (short)0, acc0, false, false);
        acc1 = __builtin_amdgcn_wmma_f32_16x16x4_f32(false, av, false, bv1,
                                                     (short)0, acc1, false, false);
    }

#pragma unroll
    for (int v = 0; v < 8; ++v) {
        const int rr = tileM + v + 8 * half;   // C/D layout: VGPR v -> M = v (+8)
        if (rr < M) {
            if (c0 < N)
                C[(size_t)rr * ldc + c0] =
                    epi_apply<EPI>(acc0[v], rr, c0, ldc, N, bias, aux, mask, maskL);
            if (c1 < N)
                C[(size_t)rr * ldc + c1] =
                    epi_apply<EPI>(acc1[v], rr, c1, ldc, N, bias, aux, mask, maskL);
        }
    }
}

// ---------------------------------------------------------------------------
// h2[0:16]  = h ; h2[16:32] = h reversed along L   (both mamba directions)
// ---------------------------------------------------------------------------
__global__ void build_h2_kernel(const float* __restrict__ h, float* __restrict__ h2)
{
    const int idx = blockIdx.x * blockDim.x + threadIdx.x;
    if (idx >= ROWS2 * DMODEL) return;
    const int d   = idx % DMODEL;
    const int rw  = idx / DMODEL;
    const int b2  = rw / LEN;
    const int l   = rw % LEN;
    const int src = (b2 < BATCH) ? (b2 * LEN + l)
                                 : ((b2 - BATCH) * LEN + (LEN - 1 - l));
    h2[idx] = h[(size_t)src * DMODEL + d];
}

// ---------------------------------------------------------------------------
// Depthwise causal conv (k=4) + bias + SiLU on xz[..., 0:DINNER]
// ---------------------------------------------------------------------------
__global__ void conv_silu_kernel(const float* __restrict__ xz,
                                 const float* __restrict__ cw,
                                 const float* __restrict__ cb,
                                 float* __restrict__ xc)
{
    const int idx = blockIdx.x * blockDim.x + threadIdx.x;
    if (idx >= ROWS2 * DINNER) return;
    const int c  = idx & (DINNER - 1);
    const int rw = idx >> 10;          // b2*LEN + l
    const int l  = rw % LEN;
    const int b  = rw / LEN;
    float s = cb[c];
#pragma unroll
    for (int k = 0; k < 4; ++k) {
        const int ll = l - 3 + k;
        if (ll >= 0)
            s += cw[c * 4 + k] * xz[(size_t)(b * LEN + ll) * (2 * DINNER) + c];
    }
    xc[idx] = silu_f(s);
}

// ---------------------------------------------------------------------------
// Selective scan: one block per sequence (b2), 1024 threads = one channel d.
// Bm/Cm (16 each) staged in LDS per timestep, shared by all 1024 channels.
// Fuses  y *= silu(z)  at the end of each step.
// ---------------------------------------------------------------------------
__global__ __launch_bounds__(1024) void scan_kernel(
    const float* __restrict__ xdbl,   // (ROWS2, 64): [dt_in | Bm | Cm]
    const float* __restrict__ dtb,    // (ROWS2, DINNER) softplus(dt)
    const float* __restrict__ xc,     // (ROWS2, DINNER)
    const float* __restrict__ xz,     // (ROWS2, 2*DINNER), z at cols [DINNER,2*DINNER)
    const float* __restrict__ A_log,  // (DINNER, DSTATE)
    const float* __restrict__ Dv,     // (DINNER,)
    float* __restrict__ ym)           // (ROWS2, DINNER)
{
    const int b = blockIdx.x;      // 0..31
    const int d = threadIdx.x;     // 0..1023
    __shared__ float sBC[2 * DSTATE];

    float a[DSTATE], hst[DSTATE];
#pragma unroll
    for (int s = 0; s < DSTATE; ++s) {
        a[s]   = -expf(A_log[d * DSTATE + s]);
        hst[s] = 0.0f;
    }
    const float Dd = Dv[d];

    for (int l = 0; l < LEN; ++l) {
        const size_t rb = (size_t)(b * LEN + l);
        if (threadIdx.x < 2 * DSTATE)
            sBC[threadIdx.x] = xdbl[rb * 64 + DTRANK + threadIdx.x];
        __syncthreads();

        const float dtv = dtb[rb * DINNER + d];
        const float xv  = xc[rb * DINNER + d];
        const float du  = dtv * xv;
        float acc = 0.0f;
#pragma unroll
        for (int s = 0; s < DSTATE; ++s) {
            hst[s] = expf(dtv * a[s]) * hst[s] + du * sBC[s];
            acc += hst[s] * sBC[DSTATE + s];
        }
        const float z = xz[rb * (2 * DINNER) + DINNER + d];
        ym[rb * DINNER + d] = (acc + xv * Dd) * silu_f(z);
        __syncthreads();
    }
}

// ---------------------------------------------------------------------------
// o = o2[fwd] + flip(o2[rev])
// ---------------------------------------------------------------------------
__global__ void combine_kernel(const float* __restrict__ o2, float* __restrict__ o)
{
    const int idx = blockIdx.x * blockDim.x + threadIdx.x;
    if (idx >= ROWS * DMODEL) return;
    const int d  = idx % DMODEL;
    const int rw = idx / DMODEL;
    const int b  = rw / LEN;
    const int l  = rw % LEN;
    const size_t rev = (size_t)((b + BATCH) * LEN + (LEN - 1 - l));
    o[idx] = o2[idx] + o2[rev * DMODEL + d];
}

// ---------------------------------------------------------------------------
// Per-batch tanh layernorm (mean/var over LEN*DMODEL)
// ---------------------------------------------------------------------------
__global__ __launch_bounds__(256) void norm_stats_kernel(const float* __restrict__ x,
                                                         float* __restrict__ stats)
{
    const int b = blockIdx.x;
    const int n = LEN * DMODEL;                 // 97280
    const float* xb = x + (size_t)b * n;
    float s = 0.0f, s2 = 0.0f;
    for (int i = threadIdx.x; i < n; i += 256) {
        const float t = tanhf(xb[i]);
        s += t; s2 += t * t;
    }
    __shared__ float rs[256], rq[256];
    rs[threadIdx.x] = s; rq[threadIdx.x] = s2;
    __syncthreads();
    for (int off = 128; off > 0; off >>= 1) {
        if (threadIdx.x < off) {
            rs[threadIdx.x] += rs[threadIdx.x + off];
            rq[threadIdx.x] += rq[threadIdx.x + off];
        }
        __syncthreads();
    }
    if (threadIdx.x == 0) {
        const float mu  = rs[0] / (float)n;
        const float var = rq[0] / (float)n - mu * mu;
        stats[b * 2 + 0] = mu;
        stats[b * 2 + 1] = rsqrtf(var + 1e-5f);
    }
}

__global__ void norm_apply_kernel(const float* __restrict__ x,
                                  const float* __restrict__ stats,
                                  const float* __restrict__ nw,
                                  const float* __restrict__ nb,
                                  float* __restrict__ out)
{
    const int idx = blockIdx.x * blockDim.x + threadIdx.x;
    if (idx >= BATCH * LEN * DMODEL) return;
    const int b  = idx / (LEN * DMODEL);
    const int ld = idx % (LEN * DMODEL);
    const float mu = stats[b * 2 + 0];
    const float rs = stats[b * 2 + 1];
    out[idx] = (tanhf(x[idx]) - mu) * rs * nw[ld] + nb[ld];
}

// ---------------------------------------------------------------------------
// Host-side orchestration
// ---------------------------------------------------------------------------
static inline dim3 gemm_grid(int M, int N) { return dim3((N + 63) / 64, (M + 31) / 32); }

extern "C" void kernel_launch(void* const* d_in, const int* in_sizes, int n_in,
                              void* d_out, int out_size, void* d_ws, size_t ws_size,
                              hipStream_t stream)
{
    (void)in_sizes; (void)n_in; (void)out_size; (void)ws_size;

    const float*         x            = (const float*)d_in[0];
    const unsigned char* pc_mask      = (const unsigned char*)d_in[1];   // bool (190,190)
    const float*         in_resize_w  = (const float*)d_in[2];
    const float*         in_resize_b  = (const float*)d_in[3];
    const float*         in_reset_w   = (const float*)d_in[4];
    const float*         in_reset_b   = (const float*)d_in[5];
    const float*         out_resize_w = (const float*)d_in[6];
    const float*         out_resize_b = (const float*)d_in[7];
    const float*         out_reset_w  = (const float*)d_in[8];
    const float*         out_reset_b  = (const float*)d_in[9];
    const float*         in_proj_w    = (const float*)d_in[10];
    const float*         conv_w       = (const float*)d_in[11];
    const float*         conv_b       = (const float*)d_in[12];
    const float*         x_proj_w     = (const float*)d_in[13];
    const float*         dt_proj_w    = (const float*)d_in[14];
    const float*         dt_proj_b    = (const float*)d_in[15];
    const float*         A_log        = (const float*)d_in[16];
    const float*         Dv           = (const float*)d_in[17];
    const float*         out_proj_w   = (const float*)d_in[18];
    const float*         norm_w       = (const float*)d_in[19];
    const float*         norm_b       = (const float*)d_in[20];

    float* ws = (float*)d_ws;
    // Workspace layout (floats); later stages reuse earlier buffers.
    float* t1    = ws;                                   // (3040,190)  -> later t2
    float* h     = t1    + (size_t)ROWS  * LEN;          // (3040,512)  -> later o
    float* h2    = h     + (size_t)ROWS  * DMODEL;       // (6080,512)  -> later o2
    float* xz    = h2    + (size_t)ROWS2 * DMODEL;       // (6080,2048)
    float* xc    = xz    + (size_t)ROWS2 * 2 * DINNER;   // (6080,1024)
    float* xdbl  = xc    + (size_t)ROWS2 * DINNER;       // (6080,64)
    float* dtb   = xdbl  + (size_t)ROWS2 * 64;           // (6080,1024)
    float* ym    = dtb   + (size_t)ROWS2 * DINNER;       // (6080,1024)
    float* stats = ym    + (size_t)ROWS2 * DINNER;       // (16,2)
    float* o2 = h2;   // reuse after xz is built
    float* o  = h;    // reuse after h2 is built
    float* t2 = t1;   // reuse

    float* out_norm = (float*)d_out;                       // output 0: (16,190,512)
    float* out_o    = out_norm + (size_t)ROWS * DMODEL;    // output 1: (16,190,512)

    const dim3 blk(128);

    // 1) t1 = x @ in_resize_w^T + b      (mask[0,0] on the diagonal == 1)
    wmma_gemm_nt<1><<<gemm_grid(ROWS, LEN), blk, 0, stream>>>(
        x, DMODEL, in_resize_w, DMODEL, t1, LEN, ROWS, LEN, DMODEL,
        in_resize_b, nullptr, nullptr, 1);
    // 2) h = (t1 @ in_reset_w^T + b) * x
    wmma_gemm_nt<2><<<gemm_grid(ROWS, DMODEL), blk, 0, stream>>>(
        t1, LEN, in_reset_w, LEN, h, DMODEL, ROWS, DMODEL, LEN,
        in_reset_b, x, nullptr, 1);
    // 3) h2 = [h ; flip(h)]
    {
        const int n = ROWS2 * DMODEL;
        build_h2_kernel<<<(n + 255) / 256, 256, 0, stream>>>(h, h2);
    }
    // 4) xz = h2 @ in_proj_w^T
    wmma_gemm_nt<0><<<gemm_grid(ROWS2, 2 * DINNER), blk, 0, stream>>>(
        h2, DMODEL, in_proj_w, DMODEL, xz, 2 * DINNER, ROWS2, 2 * DINNER, DMODEL,
        nullptr, nullptr, nullptr, 1);
    // 5) xc = silu(conv(xz[:, :DINNER]) + conv_b)
    {
        const int n = ROWS2 * DINNER;
        conv_silu_kernel<<<(n + 255) / 256, 256, 0, stream>>>(xz, conv_w, conv_b, xc);
    }
    // 6) xdbl = xc @ x_proj_w^T
    wmma_gemm_nt<0><<<gemm_grid(ROWS2, DTRANK + 2 * DSTATE), blk, 0, stream>>>(
        xc, DINNER, x_proj_w, DINNER, xdbl, 64, ROWS2, DTRANK + 2 * DSTATE, DINNER,
        nullptr, nullptr, nullptr, 1);
    // 7) dt = softplus(xdbl[:, :32] @ dt_proj_w^T + b)
    wmma_gemm_nt<4><<<gemm_grid(ROWS2, DINNER), blk, 0, stream>>>(
        xdbl, 64, dt_proj_w, DTRANK, dtb, DINNER, ROWS2, DINNER, DTRANK,
        dt_proj_b, nullptr, nullptr, 1);
    // 8) selective scan (+ fused *silu(z))
    scan_kernel<<<B2, DINNER, 0, stream>>>(xdbl, dtb, xc, xz, A_log, Dv, ym);
    // 9) o2 = ym @ out_proj_w^T
    wmma_gemm_nt<0><<<gemm_grid(ROWS2, DMODEL), blk, 0, stream>>>(
        ym, DINNER, out_proj_w, DINNER, o2, DMODEL, ROWS2, DMODEL, DINNER,
        nullptr, nullptr, nullptr, 1);
    // 10) o = o2[fwd] + flip(o2[rev])
    {
        const int n = ROWS * DMODEL;
        combine_kernel<<<(n + 255) / 256, 256, 0, stream>>>(o2, o);
    }
    // 11) t2 = (o @ out_resize_w^T + b) * mask[l, :]
    wmma_gemm_nt<3><<<gemm_grid(ROWS, LEN), blk, 0, stream>>>(
        o, DMODEL, out_resize_w, DMODEL, t2, LEN, ROWS, LEN, DMODEL,
        out_resize_b, nullptr, pc_mask, LEN);
    // 12) out_o = (t2 @ out_reset_w^T + b) * x
    wmma_gemm_nt<2><<<gemm_grid(ROWS, DMODEL), blk, 0, stream>>>(
        t2, LEN, out_reset_w, LEN, out_o, DMODEL, ROWS, DMODEL, LEN,
        out_reset_b, x, nullptr, 1);
    // 13) tanh layernorm (per batch) -> out_norm
    norm_stats_kernel<<<BATCH, 256, 0, stream>>>(x, stats);
    {
        const int n = BATCH * LEN * DMODEL;
        norm_apply_kernel<<<(n + 255) / 256, 256, 0, stream>>>(x, stats, norm_w, norm_b, out_norm);
    }
}